// SimpleGraphConv_24086176595995
// MI455X (gfx1250) — compile-verified
//
#include <hip/hip_runtime.h>
#include <hip/hip_bf16.h>

typedef __attribute__((ext_vector_type(2))) float v2f;
typedef __attribute__((ext_vector_type(8))) float v8f;

#define IN_C  128
#define OUT_C 128

// ---------------- utility kernels ----------------

__global__ __launch_bounds__(256) void rgcn_zero_u32(unsigned* p, int n) {
    int i = blockIdx.x * 256 + threadIdx.x;
    if (i < n) p[i] = 0u;
}

__global__ __launch_bounds__(256) void rgcn_init_out(float* out, const float* __restrict__ bias, int total) {
    int i = blockIdx.x * 256 + threadIdx.x;
    if (i < total) out[i] = bias[i & (OUT_C - 1)];
}

// one thread per edge: histogram edges into (relation, dst) bins
__global__ __launch_bounds__(256) void rgcn_count(const int* __restrict__ dst,
                                                  const int* __restrict__ et,
                                                  unsigned* cnt, int E, int N) {
    int e = blockIdx.x * 256 + threadIdx.x;
    if (e < E) {
        int r = et[e];
        int d = dst[e];
        atomicAdd(&cnt[(long)r * N + d], 1u);
    }
}

// in-place: u32 count -> f32 reciprocal (0 where count==0)
__global__ __launch_bounds__(256) void rgcn_inv(unsigned* cnt, int total) {
    int i = blockIdx.x * 256 + threadIdx.x;
    if (i < total) {
        unsigned c = cnt[i];
        float v = c ? (1.0f / (float)c) : 0.0f;
        ((float*)cnt)[i] = v;
    }
}

// ---------------- fp32 WMMA GEMM: T = x @ W_r ----------------
// Block = 256 threads (8 waves). Each block computes a 16x128 output slab.
// LDS stages the 16x128 x-tile once; each wave owns one 16x16 N-tile and
// issues 32 v_wmma_f32_16x16x4_f32 over K=128.
__global__ __launch_bounds__(256) void rgcn_gemm(const float* __restrict__ x,
                                                 const float* __restrict__ W,
                                                 float* __restrict__ T, int N) {
    __shared__ float sX[16 * IN_C];   // 8 KB
    const int tid = threadIdx.x;
    const int rowBase = blockIdx.x * 16;

    // cooperative load of the x tile (zero-pad past N)
    for (int idx = tid; idx < 16 * IN_C; idx += 256) {
        int r = idx >> 7;
        int c = idx & (IN_C - 1);
        int gr = rowBase + r;
        sX[idx] = (gr < N) ? x[(long)gr * IN_C + c] : 0.0f;
    }
    __syncthreads();

    const int wave = tid >> 5;
    const int lane = tid & 31;
    const int n0 = wave * 16;        // this wave's N-tile
    const int hl = lane >> 4;        // half-select: lanes 0-15 vs 16-31
    const int ll = lane & 15;

    v8f acc = {};
    for (int k0 = 0; k0 < IN_C; k0 += 4) {
        // A fragment (16x4 f32): lane<16 holds K={k0,k0+1}, lane>=16 K={k0+2,k0+3}, row M = lane&15
        const int kb = k0 + (hl << 1);
        v2f a, b;
        a.x = sX[ll * IN_C + kb];
        a.y = sX[ll * IN_C + kb + 1];
        // B fragment (4x16 f32): mirrored K split, column N = n0 + (lane&15)
        b.x = W[(long)kb * OUT_C + n0 + ll];
        b.y = W[(long)(kb + 1) * OUT_C + n0 + ll];
        acc = __builtin_amdgcn_wmma_f32_16x16x4_f32(
            /*neg_a=*/false, a, /*neg_b=*/false, b,
            /*c_mod=*/(short)0, acc, /*reuse_a=*/false, /*reuse_b=*/false);
    }

    // C/D layout: VGPR v -> row v (lanes 0-15) / row v+8 (lanes 16-31), col = lane&15
    const int cn = n0 + ll;
    const int rbase = rowBase + hl * 8;
#pragma unroll
    for (int v = 0; v < 8; ++v) {
        int gr = rbase + v;
        if (gr < N) T[(long)gr * OUT_C + cn] = acc[v];
    }
}

// ---------------- scatter: out[dst] += T[src] * inv[r][dst] ----------------
// One wave per edge; 4 channels per lane via float4 + global_atomic_add_f32.
__global__ __launch_bounds__(256) void rgcn_scatter(const int* __restrict__ src,
                                                    const int* __restrict__ dst,
                                                    const int* __restrict__ et,
                                                    const float* __restrict__ T,
                                                    const float* __restrict__ inv,
                                                    float* out, int E, int N, int rel) {
    const int wave = threadIdx.x >> 5;
    const int lane = threadIdx.x & 31;
    const long e = (long)blockIdx.x * 8 + wave;
    if (e >= E) return;
    if (et[e] != rel) return;

    const int s = src[e];
    const int d = dst[e];
    const float sc = inv[(long)rel * N + d];

    const float4* ts = (const float4*)(T + (long)s * OUT_C);
    float4 v = ts[lane];
    float* ob = out + (long)d * OUT_C + lane * 4;
#if defined(__HIP_DEVICE_COMPILE__)
    unsafeAtomicAdd(ob + 0, v.x * sc);
    unsafeAtomicAdd(ob + 1, v.y * sc);
    unsafeAtomicAdd(ob + 2, v.z * sc);
    unsafeAtomicAdd(ob + 3, v.w * sc);
#else
    atomicAdd(ob + 0, v.x * sc);
    atomicAdd(ob + 1, v.y * sc);
    atomicAdd(ob + 2, v.z * sc);
    atomicAdd(ob + 3, v.w * sc);
#endif
}

// ---------------- host-side launcher ----------------

extern "C" void kernel_launch(void* const* d_in, const int* in_sizes, int n_in,
                              void* d_out, int out_size, void* d_ws, size_t ws_size,
                              hipStream_t stream) {
    const float* x    = (const float*)d_in[0];
    const int*   ei   = (const int*)d_in[1];   // [2, E]: row0 = src, row1 = dst
    const int*   et   = (const int*)d_in[2];   // [E]
    const float* w    = (const float*)d_in[3]; // [R, 128, 128]
    const float* bias = (const float*)d_in[4]; // [128]

    const int N = in_sizes[0] / IN_C;
    const int E = in_sizes[2];
    const int R = in_sizes[3] / (IN_C * OUT_C);

    const int* src = ei;
    const int* dst = ei + E;
    float* out = (float*)d_out;

    // workspace layout: [R*N u32 counts -> f32 reciprocals][16x-aligned][N*128 f32 T]
    unsigned* cnt = (unsigned*)d_ws;
    size_t cntBytes = ((size_t)R * N * sizeof(unsigned) + 255) & ~(size_t)255;
    float* T = (float*)((char*)d_ws + cntBytes);

    const int totRN = R * N;
    const int totOut = N * OUT_C;

    rgcn_zero_u32<<<(totRN + 255) / 256, 256, 0, stream>>>(cnt, totRN);
    rgcn_init_out<<<(totOut + 255) / 256, 256, 0, stream>>>(out, bias, totOut);
    rgcn_count<<<(E + 255) / 256, 256, 0, stream>>>(dst, et, cnt, E, N);
    rgcn_inv<<<(totRN + 255) / 256, 256, 0, stream>>>(cnt, totRN);

    const float* inv = (const float*)cnt;
    const int gemmGrid = (N + 15) / 16;
    const int scatGrid = (E + 7) / 8;
    for (int r = 0; r < R; ++r) {
        rgcn_gemm<<<gemmGrid, 256, 0, stream>>>(x, w + (long)r * IN_C * OUT_C, T, N);
        rgcn_scatter<<<scatGrid, 256, 0, stream>>>(src, dst, et, T, inv, out, E, N, r);
    }
}